// LocalGatedPropagation_42356967473764
// MI455X (gfx1250) — compile-verified
//
#include <hip/hip_runtime.h>
#include <hip/hip_bf16.h>
#include <math.h>

// ---------------------------------------------------------------------------
// LocalGatedPropagation for MI455X (gfx1250, wave32)
//   q:(1,64,96,96) k:(1,64,96,96) v:(1,1024,96,96)
//   rel_w:(225,64) rel_b:(225) dw:(1024,1,5,5) proj_w:(1024,1024) proj_b:(1024)
//   out:(9216,1,1024) fp32
// Stage 5 (dominant 19.3 GFLOP GEMM) runs on v_wmma_f32_16x16x4_f32 with a
// 64x64-per-wave register-blocked tile (16 accumulators, wmma:vmem = 2:1).
// ---------------------------------------------------------------------------

#define H 96
#define W_ 96
#define HW 9216
#define CV 1024
#define DATT 64
#define WIN 15
#define NWIN 225
#define MAXD 7

typedef float v2f __attribute__((ext_vector_type(2)));
typedef float v8f __attribute__((ext_vector_type(8)));

// ---------------------------------------------------------------------------
// Kernel 1: qk[o, p] = valid ? 0.125*sum_c q[c,p]*k[c,kp] : 0
//                      + sum_c q[c,p]*rel_w[o,c] + rel_b[o]
// one block per pixel; q vector staged in LDS.
// ---------------------------------------------------------------------------
__global__ __launch_bounds__(256) void lgp_qk_kernel(
    const float* __restrict__ q, const float* __restrict__ k,
    const float* __restrict__ rel_w, const float* __restrict__ rel_b,
    float* __restrict__ qk)
{
    __shared__ float qs[DATT];
    const int p   = blockIdx.x;
    const int tid = threadIdx.x;
    const int y = p / W_, x = p % W_;

    if (tid < DATT) qs[tid] = q[tid * HW + p];
    __syncthreads();

    if (tid < NWIN) {
        const int dy = tid / WIN - MAXD;
        const int dx = tid % WIN - MAXD;
        const int ky = y + dy, kx = x + dx;
        const bool valid = ((unsigned)ky < (unsigned)H) && ((unsigned)kx < (unsigned)W_);
        const int kp = ky * W_ + kx;

        const float* __restrict__ wr = rel_w + tid * DATT;
        float rel = 0.f;
        float corr = 0.f;
        if (valid) {
            const float* __restrict__ kc = k + kp;
            #pragma unroll 8
            for (int c = 0; c < DATT; ++c) {
                rel  = fmaf(qs[c], wr[c], rel);
                corr = fmaf(qs[c], kc[c * HW], corr);
            }
        } else {
            #pragma unroll 8
            for (int c = 0; c < DATT; ++c)
                rel = fmaf(qs[c], wr[c], rel);
        }
        qk[tid * HW + p] = corr * 0.125f + rel + rel_b[tid];
    }
}

// ---------------------------------------------------------------------------
// Kernel 2: softmax over the 225 window slots per pixel (in place).
// ---------------------------------------------------------------------------
__global__ __launch_bounds__(256) void lgp_softmax_kernel(float* __restrict__ qk)
{
    __shared__ float red[256];
    const int p   = blockIdx.x;
    const int tid = threadIdx.x;

    float val = (tid < NWIN) ? qk[tid * HW + p] : -INFINITY;
    red[tid] = val;
    __syncthreads();
    #pragma unroll
    for (int s = 128; s > 0; s >>= 1) {
        if (tid < s) red[tid] = fmaxf(red[tid], red[tid + s]);
        __syncthreads();
    }
    const float m = red[0];
    __syncthreads();

    const float e = (tid < NWIN) ? __expf(val - m) : 0.f;
    red[tid] = e;
    __syncthreads();
    #pragma unroll
    for (int s = 128; s > 0; s >>= 1) {
        if (tid < s) red[tid] += red[tid + s];
        __syncthreads();
    }
    const float inv = 1.f / red[0];
    if (tid < NWIN) qk[tid * HW + p] = e * inv;
}

// ---------------------------------------------------------------------------
// Kernel 3: aggregation  agg[p, c] = sum_{valid o} attn[o,p] * v[c, key(p,o)]
// lanes stride over x (pixels) so v reads are contiguous 128B lines (v is
// channel-major); attn reads are likewise coalesced over p.
// grid (3 x-seg, 96 rows, 128 ch-groups), block (32, 8).
// ---------------------------------------------------------------------------
__global__ __launch_bounds__(256) void lgp_aggregate_kernel(
    const float* __restrict__ attn, const float* __restrict__ v,
    float* __restrict__ agg)
{
    const int x = blockIdx.x * 32 + threadIdx.x;
    const int y = blockIdx.y;
    const int c = blockIdx.z * 8 + threadIdx.y;
    const int p = y * W_ + x;

    const float* __restrict__ vc = v + (size_t)c * HW;
    float acc = 0.f;
    #pragma unroll
    for (int wy = 0; wy < WIN; ++wy) {
        const int ky = y + wy - MAXD;
        if ((unsigned)ky >= (unsigned)H) continue;
        const int rowbase = ky * W_;
        #pragma unroll
        for (int wx = 0; wx < WIN; ++wx) {
            const int kx = x + wx - MAXD;
            if ((unsigned)kx >= (unsigned)W_) continue;
            const int o = wy * WIN + wx;
            acc = fmaf(attn[o * HW + p], vc[rowbase + kx], acc);
        }
    }
    agg[(size_t)p * CV + c] = acc;
}

// ---------------------------------------------------------------------------
// Kernel 4: depthwise 5x5 conv, padding 2, no bias.
// agg/xconv are pixel-major channel-fastest -> lanes over c are coalesced.
// ---------------------------------------------------------------------------
__global__ __launch_bounds__(256) void lgp_dwconv_kernel(
    const float* __restrict__ agg, const float* __restrict__ dwk,
    float* __restrict__ xconv)
{
    const int p   = blockIdx.x;
    const int tid = threadIdx.x;
    const int y = p / W_, x = p % W_;

    for (int r = 0; r < CV / 256; ++r) {
        const int c = tid + r * 256;
        float wloc[25];
        #pragma unroll
        for (int t = 0; t < 25; ++t) wloc[t] = dwk[c * 25 + t];

        float acc = 0.f;
        #pragma unroll
        for (int i = 0; i < 5; ++i) {
            const int py = y + i - 2;
            if ((unsigned)py >= (unsigned)H) continue;
            #pragma unroll
            for (int j = 0; j < 5; ++j) {
                const int px = x + j - 2;
                if ((unsigned)px >= (unsigned)W_) continue;
                acc = fmaf(wloc[i * 5 + j],
                           agg[(size_t)(py * W_ + px) * CV + c], acc);
            }
        }
        xconv[(size_t)p * CV + c] = acc;
    }
}

// ---------------------------------------------------------------------------
// Kernel 5: projection GEMM  out[p, co] = sum_ci X[p,ci]*Wm[co,ci] + b[co]
// fp32 WMMA: D(16x16) += A(16x4) x B(4x16), 256 K-steps.
// A-frag (ISA 7.12.2, 32-bit A 16x4): lanes 0-15 hold K={0,1}, lanes 16-31
// hold K={2,3}, M = lane%16  -> one float2 load from X row.
// B-frag mirrors it over N: B[k][n] = Wm[n][k] -> float2 load from Wm row.
// Register blocking: one wave owns a 64x64 output tile = 4x4 sub-tiles of
// 16x16 (16 v8f accumulators = 128 VGPRs). Per K-step: 8 b64 loads feed
// 16 v_wmma ops -> wmma:vmem = 2:1 and 4x less redundant operand traffic.
// grid (16 n-tiles, 144 m-tiles), block 32 (one wave).
// ---------------------------------------------------------------------------
__global__ __launch_bounds__(32) void lgp_proj_wmma_kernel(
    const float* __restrict__ X, const float* __restrict__ Wm,
    const float* __restrict__ bias, float* __restrict__ out)
{
    const int lane = threadIdx.x;
    const int l    = lane & 15;
    const int half = lane >> 4;

    const int tn = blockIdx.x * 64;     // output-channel tile base (0..960)
    const int tm = blockIdx.y * 64;     // pixel tile base          (0..9152)

    const float* __restrict__ arow[4];
    const float* __restrict__ brow[4];
    #pragma unroll
    for (int i = 0; i < 4; ++i) {
        arow[i] = X  + (size_t)(tm + i * 16 + l) * CV + 2 * half;
        brow[i] = Wm + (size_t)(tn + i * 16 + l) * CV + 2 * half;
    }

    v8f acc[4][4];
    #pragma unroll
    for (int i = 0; i < 4; ++i)
        #pragma unroll
        for (int j = 0; j < 4; ++j)
            acc[i][j] = (v8f){};

    for (int k0 = 0; k0 < CV; k0 += 4) {
        v2f a[4], b[4];
        #pragma unroll
        for (int i = 0; i < 4; ++i) a[i] = *(const v2f*)(arow[i] + k0);
        #pragma unroll
        for (int j = 0; j < 4; ++j) b[j] = *(const v2f*)(brow[j] + k0);

        #pragma unroll
        for (int i = 0; i < 4; ++i)
            #pragma unroll
            for (int j = 0; j < 4; ++j)
                acc[i][j] = __builtin_amdgcn_wmma_f32_16x16x4_f32(
                    /*neg_a=*/false, a[i], /*neg_b=*/false, b[j],
                    /*c_mod=*/(short)0, acc[i][j],
                    /*reuse_a=*/false, /*reuse_b=*/false);
    }

    #pragma unroll
    for (int j = 0; j < 4; ++j) {
        const int col  = tn + j * 16 + l;
        const float bb = bias[col];
        #pragma unroll
        for (int i = 0; i < 4; ++i) {
            const int rowbase = tm + i * 16 + 8 * half;  // D: lanes16-31 -> M+8
            #pragma unroll
            for (int r = 0; r < 8; ++r)
                out[(size_t)(rowbase + r) * CV + col] = acc[i][j][r] + bb;
        }
    }
}

// ---------------------------------------------------------------------------
extern "C" void kernel_launch(void* const* d_in, const int* in_sizes, int n_in,
                              void* d_out, int out_size, void* d_ws, size_t ws_size,
                              hipStream_t stream)
{
    const float* q      = (const float*)d_in[0];
    const float* k      = (const float*)d_in[1];
    const float* v      = (const float*)d_in[2];
    const float* rel_w  = (const float*)d_in[3];
    const float* rel_b  = (const float*)d_in[4];
    const float* dwk    = (const float*)d_in[5];
    const float* proj_w = (const float*)d_in[6];
    const float* proj_b = (const float*)d_in[7];
    float* out = (float*)d_out;

    float* ws    = (float*)d_ws;
    float* qk    = ws;                          // 225 * 9216        =  2,073,600 f
    float* agg   = qk  + (size_t)NWIN * HW;     // 9216 * 1024       =  9,437,184 f
    float* xconv = agg + (size_t)HW * CV;       // 9216 * 1024       =  9,437,184 f
    // total scratch: ~83.8 MB (fits the preallocated ws / L2)

    lgp_qk_kernel<<<HW, 256, 0, stream>>>(q, k, rel_w, rel_b, qk);
    lgp_softmax_kernel<<<HW, 256, 0, stream>>>(qk);
    lgp_aggregate_kernel<<<dim3(3, H, CV / 8), dim3(32, 8), 0, stream>>>(qk, v, agg);
    lgp_dwconv_kernel<<<HW, 256, 0, stream>>>(agg, dwk, xconv);
    lgp_proj_wmma_kernel<<<dim3(CV / 64, HW / 64), 32, 0, stream>>>(xconv, proj_w, proj_b, out);
}